// SOM_11940009083349
// MI455X (gfx1250) — compile-verified
//
#include <hip/hip_runtime.h>

typedef __attribute__((ext_vector_type(2))) float v2f;
typedef __attribute__((ext_vector_type(8))) float v8f;
typedef __attribute__((ext_vector_type(4))) int   v4i;

#define SOM_N      500000
#define SOM_TILES  (SOM_N / 16)   // 31250, exact
#define WAVES_PER_BLOCK 8

__global__ __launch_bounds__(256) void
som_bmu_wmma_kernel(const float* __restrict__ x,
                    const float* __restrict__ w,   // [16][16][16]
                    int* __restrict__ out)         // [N][16]
{
    const int lane = threadIdx.x & 31;
    const int wave = threadIdx.x >> 5;
    const int half = lane >> 4;     // 0: lanes 0-15, 1: lanes 16-31
    const int l15  = lane & 15;

    // ---------------------------------------------------------------
    // Transposed GEMM: D[m1][xrow] = A x B,
    //   A = [-2*W[i0] | w2[i0]]  (16 m1-rows x K=20)
    //   B = [x-tile^T | ones]    (K=20 x 16 x-rows)
    // A/B fragments share the same per-lane striping:
    //   element e of fragment f holds K = 4f + 2*half + e,
    //   M (A) / N (B) = lane & 15.
    // Register-resident A for all 16 m0 groups; -2 folded in once.
    // ---------------------------------------------------------------
    v2f   af[16][4];
    float w2v[16];   // A frag4: lane<16 -> {w2, 0}; lane>=16 -> {0, 0}
#pragma unroll
    for (int i0 = 0; i0 < 16; ++i0) {
        const float* wr = w + (i0 * 16 + l15) * 16;   // weights row m1 = l15
        float part = 0.0f;
#pragma unroll
        for (int f = 0; f < 4; ++f) {
            const int kb = 4 * f + 2 * half;
            const float w0 = wr[kb], w1 = wr[kb + 1];
            part = fmaf(w0, w0, part);
            part = fmaf(w1, w1, part);
            v2f a; a.x = -2.0f * w0; a.y = -2.0f * w1;
            af[i0][f] = a;
        }
        const float w2full = part + __shfl_xor(part, 16, 32);
        w2v[i0] = half ? 0.0f : w2full;   // K=17..19 rows are zero
    }
    // B frag4: K=16 row of B is all-ones (multiplies w2), K=17..19 zero.
    v2f b4; b4.x = half ? 0.0f : 1.0f; b4.y = 0.0f;

    const int waveGlobal = blockIdx.x * WAVES_PER_BLOCK + wave;
    const int waveStride = gridDim.x * WAVES_PER_BLOCK;

    for (int tile = waveGlobal; tile < SOM_TILES; tile += waveStride) {
        const int rowBase = tile * 16;

        // B fragments: x row N = l15 of this tile (raw loads, no scaling).
        const float* xr = x + (size_t)(rowBase + l15) * 16;
        v2f bx[4];
#pragma unroll
        for (int f = 0; f < 4; ++f) {
            const int kb = 4 * f + 2 * half;
            bx[f].x = xr[kb];
            bx[f].y = xr[kb + 1];
        }

        int res[16];
#pragma unroll
        for (int i0 = 0; i0 < 16; ++i0) {
            v2f a4; a4.x = w2v[i0]; a4.y = 0.0f;
            v8f acc = {};
            acc = __builtin_amdgcn_wmma_f32_16x16x4_f32(false, a4,          false, b4,    (short)0, acc, false, false);
            acc = __builtin_amdgcn_wmma_f32_16x16x4_f32(false, af[i0][0],   false, bx[0], (short)0, acc, false, false);
            acc = __builtin_amdgcn_wmma_f32_16x16x4_f32(false, af[i0][1],   false, bx[1], (short)0, acc, false, false);
            acc = __builtin_amdgcn_wmma_f32_16x16x4_f32(false, af[i0][2],   false, bx[2], (short)0, acc, false, false);
            acc = __builtin_amdgcn_wmma_f32_16x16x4_f32(false, af[i0][3],   false, bx[3], (short)0, acc, false, false);

            // acc[r] = dist(m1 = r + 8*half, xrow = l15)  (+ const ||x||^2 dropped)
            // In-lane argmin over r (first-index tie-break: strict <, increasing r).
            float best = acc[0];
            int   ridx = 0;
#pragma unroll
            for (int r = 1; r < 8; ++r) {
                const int t = acc[r] < best;        // branchless -> cndmask
                best = t ? acc[r] : best;
                ridx = t ? r      : ridx;
            }
            int idx = ridx + 8 * half;

            // One cross-half combine: half0 owns m1 0..7, half1 owns 8..15.
            const float ov = __shfl_xor(best, 16, 32);
            const int   oi = __shfl_xor(idx,  16, 32);
            // lower-index wins ties: half0 keeps on tie, half1 yields on tie
            const int t = ((int)(ov < best)) | (half & (int)(ov == best));
            idx = t ? oi : idx;
            res[i0] = idx;
        }

        // Every lane now holds the full result row for xrow = l15.
        // Lanes 0-15 store 64B each -> contiguous 1KB per tile.
        if (half == 0) {
            v4i* o = (v4i*)(out + (size_t)(rowBase + l15) * 16);
            v4i p0 = { res[0],  res[1],  res[2],  res[3]  };
            v4i p1 = { res[4],  res[5],  res[6],  res[7]  };
            v4i p2 = { res[8],  res[9],  res[10], res[11] };
            v4i p3 = { res[12], res[13], res[14], res[15] };
            o[0] = p0; o[1] = p1; o[2] = p2; o[3] = p3;
        }
    }
}

extern "C" void kernel_launch(void* const* d_in, const int* in_sizes, int n_in,
                              void* d_out, int out_size, void* d_ws, size_t ws_size,
                              hipStream_t stream) {
    (void)in_sizes; (void)n_in; (void)d_ws; (void)ws_size; (void)out_size;
    const float* x = (const float*)d_in[0];   // [500000, 16] f32
    const float* w = (const float*)d_in[1];   // [16, 16, 16] f32
    int* out = (int*)d_out;                   // [500000, 16] int32

    // Persistent grid: 1024 blocks x 8 waves = 8192 waves, ~3.8 tiles each.
    dim3 grid(1024), block(256);
    som_bmu_wmma_kernel<<<grid, block, 0, stream>>>(x, w, out);
}